// GCN_easy_17008070492798
// MI455X (gfx1250) — compile-verified
//
#include <hip/hip_runtime.h>
#include <hip/hip_bf16.h>

typedef __attribute__((ext_vector_type(2))) float v2f;
typedef __attribute__((ext_vector_type(8))) float v8f;

#define GCN_NFEAT 128
#define GCN_HID   16
#define GCN_OUT   2
#define GCN_NGRAPH 64

// ---------------- init: deg = 1.0 (self loop), pool accumulators = 0 ----------------
__global__ void gcn_init_kernel(float* __restrict__ deg, float* __restrict__ pool, int N) {
    int i = blockIdx.x * blockDim.x + threadIdx.x;
    if (i < N) deg[i] = 1.0f;                       // self-loop contributes 1 to every degree
    if (i < GCN_NGRAPH * 3) pool[i] = 0.0f;         // 128 sums + 64 counts
}

// ---------------- degree accumulation over edges ----------------
__global__ void gcn_degree_kernel(const int* __restrict__ ei, float* __restrict__ deg, int E) {
    int e = blockIdx.x * blockDim.x + threadIdx.x;
    if (e >= E) return;
    int d = ei[E + e];                              // dst row of edge_index [2,E]
    atomicAdd(&deg[d], 1.0f);
}

// ---------------- dinv = rsqrt(deg) ----------------
__global__ void gcn_dinv_kernel(const float* __restrict__ deg, float* __restrict__ dinv, int N) {
    int i = blockIdx.x * blockDim.x + threadIdx.x;
    if (i >= N) return;
    float d = deg[i];
    dinv[i] = d > 0.0f ? __frsqrt_rn(d) : 0.0f;     // deg >= 1 always (self loops)
}

// ---------------- GEMM1 via WMMA f32 16x16x4: h1 = x @ W1 ----------------
// One wave (32 lanes) computes one 16x16 output tile (M=16 rows, N=16 hidden).
// Also writes the self-loop contribution h1[i]*dinv[i]^2 into agg1 (plain store).
__global__ __launch_bounds__(256) void gcn_gemm1_wmma_kernel(
        const float* __restrict__ x, const float* __restrict__ W1,
        const float* __restrict__ dinv,
        float* __restrict__ h1, float* __restrict__ agg1, int nTiles) {
    __shared__ float sW[GCN_NFEAT * GCN_HID];       // 128x16 fp32 = 8KB
    for (int i = threadIdx.x; i < GCN_NFEAT * GCN_HID; i += blockDim.x) sW[i] = W1[i];
    __syncthreads();

    int wave = threadIdx.x >> 5;
    int lane = threadIdx.x & 31;
    int tile = blockIdx.x * 8 + wave;
    if (tile >= nTiles) return;

    int mn   = lane & 15;                           // M for A loads, N for B/C/D
    int khalf = lane >> 4;                          // 0: K={0,1}, 1: K={2,3} within 16x16x4 step
    const float* arow = x + (size_t)(tile * 16 + mn) * GCN_NFEAT + 2 * khalf;

    v8f acc = {};
    #pragma unroll
    for (int k0 = 0; k0 < GCN_NFEAT; k0 += 4) {
        // A 16x4 fp32 layout: lanes 0-15 hold K={k0,k0+1}, lanes 16-31 hold K={k0+2,k0+3}
        float2 av = *(const float2*)(arow + k0);
        v2f a; a.x = av.x; a.y = av.y;
        // B 4x16 fp32 layout (mirror of A, K across VGPR/half-wave, N across lanes)
        v2f b;
        b.x = sW[(k0 + 2 * khalf + 0) * GCN_HID + mn];
        b.y = sW[(k0 + 2 * khalf + 1) * GCN_HID + mn];
        acc = __builtin_amdgcn_wmma_f32_16x16x4_f32(
                  /*neg_a=*/false, a, /*neg_b=*/false, b,
                  /*c_mod=*/(short)0, acc, /*reuse_a=*/false, /*reuse_b=*/false);
    }

    // C/D layout: VGPR r, lanes 0-15 -> row r, lanes 16-31 -> row r+8; col = lane&15
    #pragma unroll
    for (int r = 0; r < 8; ++r) {
        int row = tile * 16 + r + 8 * khalf;
        float dv = acc[r];
        float di = dinv[row];
        h1[(size_t)row * GCN_HID + mn]   = dv;
        agg1[(size_t)row * GCN_HID + mn] = dv * di * di;   // self-loop term (covers whole array)
    }
}

// ---------------- edge scatter, 16 features: agg1[dst] += h1[src]*norm ----------------
__global__ void gcn_edge16_kernel(const int* __restrict__ ei, const float* __restrict__ dinv,
                                  const float* __restrict__ h1, float* __restrict__ agg1, int E) {
    int e = blockIdx.x * blockDim.x + threadIdx.x;
    if (e >= E) return;
    int s = ei[e];
    int d = ei[E + e];
    float nrm = dinv[s] * dinv[d];
    const float4* hs = (const float4*)(h1 + (size_t)s * GCN_HID);
    float* ad = agg1 + (size_t)d * GCN_HID;
    #pragma unroll
    for (int c = 0; c < 4; ++c) {
        float4 v = hs[c];
        atomicAdd(ad + 4 * c + 0, v.x * nrm);
        atomicAdd(ad + 4 * c + 1, v.y * nrm);
        atomicAdd(ad + 4 * c + 2, v.z * nrm);
        atomicAdd(ad + 4 * c + 3, v.w * nrm);
    }
}

// ---------------- +b1, PReLU, tiny GEMM2 (16x2), self-loop into agg2 ----------------
__global__ void gcn_prelu_gemm2_kernel(const float* __restrict__ agg1,
                                       const float* __restrict__ b1,
                                       const float* __restrict__ prelu_a,
                                       const float* __restrict__ W2,
                                       const float* __restrict__ dinv,
                                       float* __restrict__ h2, float* __restrict__ agg2, int N) {
    int i = blockIdx.x * blockDim.x + threadIdx.x;
    if (i >= N) return;
    float slope = prelu_a[0];
    const float* row = agg1 + (size_t)i * GCN_HID;
    float o0 = 0.0f, o1 = 0.0f;
    #pragma unroll
    for (int k = 0; k < GCN_HID; ++k) {
        float v = row[k] + b1[k];
        v = v >= 0.0f ? v : slope * v;
        o0 = fmaf(v, W2[k * GCN_OUT + 0], o0);
        o1 = fmaf(v, W2[k * GCN_OUT + 1], o1);
    }
    h2[(size_t)i * GCN_OUT + 0] = o0;
    h2[(size_t)i * GCN_OUT + 1] = o1;
    float di = dinv[i];
    float n2 = di * di;
    agg2[(size_t)i * GCN_OUT + 0] = o0 * n2;        // self-loop term, plain store
    agg2[(size_t)i * GCN_OUT + 1] = o1 * n2;
}

// ---------------- edge scatter, 2 features ----------------
__global__ void gcn_edge2_kernel(const int* __restrict__ ei, const float* __restrict__ dinv,
                                 const float* __restrict__ h2, float* __restrict__ agg2, int E) {
    int e = blockIdx.x * blockDim.x + threadIdx.x;
    if (e >= E) return;
    int s = ei[e];
    int d = ei[E + e];
    float nrm = dinv[s] * dinv[d];
    float2 v = *(const float2*)(h2 + (size_t)s * GCN_OUT);
    atomicAdd(agg2 + (size_t)d * GCN_OUT + 0, v.x * nrm);
    atomicAdd(agg2 + (size_t)d * GCN_OUT + 1, v.y * nrm);
}

// ---------------- pooling with per-block LDS pre-reduction ----------------
__global__ __launch_bounds__(256) void gcn_pool_kernel(const float* __restrict__ agg2,
                                                       const int* __restrict__ batch,
                                                       float* __restrict__ pool, int N) {
    __shared__ float ls[GCN_NGRAPH * 3];            // [0,128): sums, [128,192): counts
    for (int i = threadIdx.x; i < GCN_NGRAPH * 3; i += blockDim.x) ls[i] = 0.0f;
    __syncthreads();
    int i = blockIdx.x * blockDim.x + threadIdx.x;
    if (i < N) {
        int g = batch[i];
        atomicAdd(&ls[g * 2 + 0], agg2[(size_t)i * GCN_OUT + 0]);
        atomicAdd(&ls[g * 2 + 1], agg2[(size_t)i * GCN_OUT + 1]);
        atomicAdd(&ls[GCN_NGRAPH * 2 + g], 1.0f);
    }
    __syncthreads();
    for (int j = threadIdx.x; j < GCN_NGRAPH * 3; j += blockDim.x) {
        float v = ls[j];
        if (v != 0.0f) atomicAdd(&pool[j], v);
    }
}

// ---------------- finalize: out = sum / max(cnt,1) + b2 ----------------
__global__ void gcn_finalize_kernel(const float* __restrict__ pool,
                                    const float* __restrict__ b2, float* __restrict__ out) {
    int i = threadIdx.x;
    if (i < GCN_NGRAPH * GCN_OUT) {
        int g = i >> 1, o = i & 1;
        float c = pool[GCN_NGRAPH * 2 + g];
        out[i] = pool[i] / fmaxf(c, 1.0f) + b2[o];
    }
}

extern "C" void kernel_launch(void* const* d_in, const int* in_sizes, int n_in,
                              void* d_out, int out_size, void* d_ws, size_t ws_size,
                              hipStream_t stream) {
    const float* x       = (const float*)d_in[0];
    const float* W1      = (const float*)d_in[1];
    const float* b1      = (const float*)d_in[2];
    const float* prelu_a = (const float*)d_in[3];
    const float* W2      = (const float*)d_in[4];
    const float* b2      = (const float*)d_in[5];
    const int*   ei      = (const int*)d_in[6];     // [2, E]
    const int*   batch   = (const int*)d_in[7];

    const int N = in_sizes[0] / GCN_NFEAT;          // 100000
    const int E = in_sizes[6] / 2;                  // 6400000
    float* out = (float*)d_out;

    // workspace layout (floats)
    float* ws   = (float*)d_ws;
    float* deg  = ws;                               // N
    float* dinv = deg  + N;                         // N
    float* h1   = dinv + N;                         // 16N
    float* agg1 = h1   + (size_t)16 * N;            // 16N
    float* h2   = agg1 + (size_t)16 * N;            // 2N
    float* agg2 = h2   + (size_t)2  * N;            // 2N
    float* pool = agg2 + (size_t)2  * N;            // 192

    const int B = 256;
    int gN = (N + B - 1) / B;
    int gE = (E + B - 1) / B;
    int nTiles = (N + 15) / 16;
    int gT = (nTiles + 7) / 8;                      // 8 waves (tiles) per 256-thread block

    gcn_init_kernel<<<gN, B, 0, stream>>>(deg, pool, N);
    gcn_degree_kernel<<<gE, B, 0, stream>>>(ei, deg, E);
    gcn_dinv_kernel<<<gN, B, 0, stream>>>(deg, dinv, N);
    gcn_gemm1_wmma_kernel<<<gT, B, 0, stream>>>(x, W1, dinv, h1, agg1, nTiles);
    gcn_edge16_kernel<<<gE, B, 0, stream>>>(ei, dinv, h1, agg1, E);
    gcn_prelu_gemm2_kernel<<<gN, B, 0, stream>>>(agg1, b1, prelu_a, W2, dinv, h2, agg2, N);
    gcn_edge2_kernel<<<gE, B, 0, stream>>>(ei, dinv, h2, agg2, E);
    gcn_pool_kernel<<<gN, B, 0, stream>>>(agg2, batch, pool, N);
    gcn_finalize_kernel<<<1, 128, 0, stream>>>(pool, b2, out);
}